// ConvG_4320737100492
// MI455X (gfx1250) — compile-verified
//
#include <hip/hip_runtime.h>
#include <hip/hip_bf16.h>
#include <math.h>

// ---------------------------------------------------------------------------
// Problem constants (from the reference)
// ---------------------------------------------------------------------------
#define BATCH   128
#define N0      256
#define EDG     2048
#define F_IN    128
#define H1D     256
#define H2D     128
#define NCLS    10
#define KP1     205   // ceil(0.8*256)
#define KP2     164   // ceil(0.8*205)
#define KP3     132   // ceil(0.8*164)
#define BE      (BATCH * EDG)   // 262144 edges total
#define MAXK    512             // largest GEMM K (MLP head: 2*H1D)

typedef float v2f __attribute__((ext_vector_type(2)));
typedef float v8f __attribute__((ext_vector_type(8)));
typedef int   v4i __attribute__((ext_vector_type(4)));

#if defined(__HIP_DEVICE_COMPILE__) && defined(__gfx1250__) && \
    __has_builtin(__builtin_amdgcn_global_load_async_to_lds_b128)
#define HAVE_ASYNC_LDS 1
#else
#define HAVE_ASYNC_LDS 0
#endif

#if defined(__HIP_DEVICE_COMPILE__) && defined(__gfx1250__) && \
    __has_builtin(__builtin_amdgcn_s_wait_asynccnt)
#define WAIT_ASYNC() __builtin_amdgcn_s_wait_asynccnt(0)
#elif defined(__HIP_DEVICE_COMPILE__) && defined(__gfx1250__)
#define WAIT_ASYNC() asm volatile("s_wait_asynccnt 0x0" ::: "memory")
#else
#define WAIT_ASYNC()
#endif

// ---------------------------------------------------------------------------
// fp32 WMMA GEMM:  C = act(A[M,K] @ W[K,N] + bias[N])
// Requirements: M%128==0, N%16==0, K%4==0, K<=MAXK (all GEMMs here qualify).
// One block (8 waves) = one 128x16 output strip sharing one K x 16 weight
// panel staged in LDS (async-to-LDS when available). Each wave runs
// V_WMMA_F32_16X16X4_F32 over K in steps of 4 with B operands from LDS.
// ---------------------------------------------------------------------------
__global__ __launch_bounds__(256)
void gemm_bias_act_wmma(const float* __restrict__ A, const float* __restrict__ W,
                        const float* __restrict__ bias, float* __restrict__ Cmat,
                        int M, int N, int K, int do_relu) {
  __shared__ float sW[MAXK * 16];   // K x 16 weight panel, row-major (32 KB max)

  const int ntn = N >> 4;
  const int strip = blockIdx.x;       // (M/128)*(N/16) strips total, exact grid
  const int ms = strip / ntn;         // 128-row strip index
  const int ns = strip - ms * ntn;    // 16-col panel index
  const int t  = threadIdx.x;

  // ---- stage W[:, ns*16 .. ns*16+15] into LDS ----
  const float* Wpanel = W + (size_t)(ns << 4);
#if HAVE_ASYNC_LDS
  {
    const int chunks = K << 2;        // K*16 floats / 4 floats per b128 chunk
    for (int c = t; c < chunks; c += 256) {
      const int row = c >> 2;
      const int cg  = (c & 3) << 2;   // starting column of this 4-float group
      const float* g = Wpanel + (size_t)row * N + cg;
      __builtin_amdgcn_global_load_async_to_lds_b128(
          (__attribute__((address_space(1))) v4i*)g,
          (__attribute__((address_space(3))) v4i*)(sW + (c << 2)),
          0, 0);
    }
    WAIT_ASYNC();                     // my async loads complete
  }
  __syncthreads();                    // everyone's async loads complete
#else
  for (int c = t; c < (K << 4); c += 256) {
    const int row = c >> 4;
    const int col = c & 15;
    sW[c] = Wpanel[(size_t)row * N + col];
  }
  __syncthreads();
#endif

  // ---- per-wave 16x16 tile ----
  const int wv   = t >> 5;            // wave 0..7 -> row tile inside strip
  const int lane = t & 31;
  const int half = lane >> 4;         // 0: K pair {0,1}, 1: K pair {2,3}
  const int l16  = lane & 15;

  // A: lane holds row (ms*128 + wv*16 + l16), K slots (2*half, 2*half+1)
  const float* Ap = A + (size_t)((ms << 7) + (wv << 4) + l16) * K + (half << 1);
  // B from LDS: lane holds col l16, K rows (2*half, 2*half+1)
  const float* Bl = sW + ((half << 1) << 4) + l16;

  v8f acc = {0.f, 0.f, 0.f, 0.f, 0.f, 0.f, 0.f, 0.f};
  for (int kk = 0; kk < K; kk += 4) {
    v2f a;  a.x = Ap[0];  a.y = Ap[1];
    v2f b;  b.x = Bl[0];  b.y = Bl[16];
    acc = __builtin_amdgcn_wmma_f32_16x16x4_f32(
        /*neg_a=*/false, a, /*neg_b=*/false, b,
        /*c_mod=*/(short)0, acc, /*reuse_a=*/false, /*reuse_b=*/false);
    Ap += 4;
    Bl += 64;                          // 4 rows * 16 cols
  }

  // C/D layout: VGPR r, lane -> M = base + r + 8*half, N = ns*16 + l16
  const int cn = (ns << 4) + l16;
  const float bv = bias[cn];
  float* Cp = Cmat + (size_t)((ms << 7) + (wv << 4) + (half << 3)) * N + cn;
#pragma unroll
  for (int r = 0; r < 8; ++r) {
    float v = acc[r] + bv;
    if (do_relu) v = fmaxf(v, 0.f);
    Cp[(size_t)r * N] = v;
  }
}

// ---------------------------------------------------------------------------
// Small supporting kernels
// ---------------------------------------------------------------------------
__global__ void fill_f32_kernel(float* p, float v, int n) {
  int i = blockIdx.x * blockDim.x + threadIdx.x;
  if (i < n) p[i] = v;
}

__global__ void split_edges_kernel(const int* __restrict__ edges,
                                   int* __restrict__ src, int* __restrict__ dst,
                                   float* __restrict__ m, int total) {
  int ge = blockIdx.x * blockDim.x + threadIdx.x;
  if (ge >= total) return;
  src[ge] = edges[2 * ge + 0];
  dst[ge] = edges[2 * ge + 1];
  m[ge] = 1.0f;
}

__global__ void deg_accum_kernel(const int* __restrict__ dst, const float* __restrict__ m,
                                 float* __restrict__ deg, int n, int total) {
  int ge = blockIdx.x * blockDim.x + threadIdx.x;
  if (ge >= total) return;
  float w = m[ge];
  if (w != 0.f) {
    int b = ge / EDG;
    atomicAdd(&deg[b * n + dst[ge]], w);
  }
}

__global__ void dinv_kernel(float* __restrict__ deg, int n) {
  int i = blockIdx.x * blockDim.x + threadIdx.x;
  if (i < n) deg[i] = 1.0f / sqrtf(deg[i]);   // deg >= 1, safe
}

__global__ void edge_norm_kernel(const int* __restrict__ src, const int* __restrict__ dst,
                                 const float* __restrict__ m, const float* __restrict__ dinv,
                                 float* __restrict__ enorm, int n, int total) {
  int ge = blockIdx.x * blockDim.x + threadIdx.x;
  if (ge >= total) return;
  int b = ge / EDG;
  enorm[ge] = dinv[b * n + src[ge]] * dinv[b * n + dst[ge]] * m[ge];
}

// out[i,f] = h[i,f] * dinv[i]^2   (initializes the hop output)
__global__ void scale_self_kernel(const float* __restrict__ h, const float* __restrict__ dinv,
                                  float* __restrict__ out, int rows) {
  int id = blockIdx.x * blockDim.x + threadIdx.x;
  if (id >= rows * H1D) return;
  int i = id / H1D;
  float d = dinv[i];
  out[id] = h[id] * (d * d);
}

// out[dst,f] += h[src,f] * enorm[e]
__global__ void edge_scatter_kernel(const float* __restrict__ h, const int* __restrict__ src,
                                    const int* __restrict__ dst, const float* __restrict__ enorm,
                                    float* __restrict__ out, int n) {
  int id = blockIdx.x * blockDim.x + threadIdx.x;   // total = BE * H1D = 67,108,864
  int ge = id >> 8;                                  // / 256 (H1D)
  int f  = id & 255;
  float w = enorm[ge];
  if (w == 0.f) return;
  int b = ge / EDG;
  int sg = b * n + src[ge];
  int dg = b * n + dst[ge];
  atomicAdd(out + (size_t)dg * H1D + f, h[(size_t)sg * H1D + f] * w);
}

__global__ void pw_invnorm_kernel(const float* __restrict__ pw, float* __restrict__ out) {
  __shared__ float sm[256];
  int t = threadIdx.x;
  float v = pw[t];
  sm[t] = v * v;
  __syncthreads();
  for (int off = 128; off >= 1; off >>= 1) {
    if (t < off) sm[t] += sm[t + off];
    __syncthreads();
  }
  if (t == 0) out[0] = 1.0f / sqrtf(sm[0]);
}

// score[row] = dot(h[row,:], pw) * invnorm   -- one wave32 per row
__global__ void row_dot_kernel(const float* __restrict__ h, const float* __restrict__ pw,
                               const float* __restrict__ invn, float* __restrict__ score,
                               int rows) {
  int wave = blockIdx.x * (blockDim.x >> 5) + (threadIdx.x >> 5);
  int lane = threadIdx.x & 31;
  if (wave >= rows) return;
  const float* r = h + (size_t)wave * H1D;
  float s = 0.f;
  for (int f = lane; f < H1D; f += 32) s += r[f] * pw[f];
  for (int d = 16; d >= 1; d >>= 1) s += __shfl_xor(s, d, 32);
  if (lane == 0) score[wave] = s * invn[0];
}

// Per-graph top-k (descending, ties -> lower index), iterative LDS tournament.
__global__ void topk_kernel(const float* __restrict__ score, float* __restrict__ vals,
                            int* __restrict__ perm, int n, int k) {
  __shared__ float s[256];
  __shared__ float rv[256];
  __shared__ int   ri[256];
  int b = blockIdx.x, t = threadIdx.x;
  s[t] = (t < n) ? score[b * n + t] : -3.402823466e38f;
  __syncthreads();
  for (int j = 0; j < k; ++j) {
    rv[t] = s[t]; ri[t] = t;
    __syncthreads();
    for (int off = 128; off >= 1; off >>= 1) {
      if (t < off) {
        float ov = rv[t + off]; int oi = ri[t + off];
        if (ov > rv[t] || (ov == rv[t] && oi < ri[t])) { rv[t] = ov; ri[t] = oi; }
      }
      __syncthreads();
    }
    if (t == 0) {
      vals[b * k + j] = rv[0];
      perm[b * k + j] = ri[0];
      s[ri[0]] = -3.402823466e38f;
    }
    __syncthreads();
  }
}

// hout[b,j,:] = hin[b,perm,:] * tanh(val);  inv[b,perm] = j
__global__ void pool_gather_kernel(const float* __restrict__ hin, const float* __restrict__ vals,
                                   const int* __restrict__ perm, int* __restrict__ inv,
                                   float* __restrict__ hout, int n, int k) {
  int bj = blockIdx.x;            // b*k + j
  int b = bj / k;
  int j = bj - b * k;
  int idx = perm[bj];
  float tv = tanhf(vals[bj]);
  const float* s = hin + (size_t)(b * n + idx) * H1D;
  float* d = hout + (size_t)bj * H1D;
  for (int f = threadIdx.x; f < H1D; f += blockDim.x) d[f] = s[f] * tv;
  if (threadIdx.x == 0) inv[b * n + idx] = j;
}

__global__ void edge_remap_kernel(const int* __restrict__ src, const int* __restrict__ dst,
                                  const float* __restrict__ m, const int* __restrict__ inv,
                                  int n, int* __restrict__ nsrc, int* __restrict__ ndst,
                                  float* __restrict__ nm, int total) {
  int ge = blockIdx.x * blockDim.x + threadIdx.x;
  if (ge >= total) return;
  int b = ge / EDG;
  int ns = inv[b * n + src[ge]];
  int nd = inv[b * n + dst[ge]];
  float mm = m[ge] * ((ns >= 0) ? 1.f : 0.f) * ((nd >= 0) ? 1.f : 0.f);
  nsrc[ge] = ns > 0 ? ns : 0;
  ndst[ge] = nd > 0 ? nd : 0;
  nm[ge] = mm;
}

// z[b, 0:256]   += max over nodes
// z[b, 256:512] += mean over nodes
__global__ void readout_add_kernel(const float* __restrict__ h, float* __restrict__ z, int n) {
  int b = blockIdx.x, f = threadIdx.x;
  const float* p = h + (size_t)b * n * H1D + f;
  float mx = -3.402823466e38f, sm = 0.f;
  for (int i = 0; i < n; ++i) {
    float v = p[(size_t)i * H1D];
    mx = fmaxf(mx, v);
    sm += v;
  }
  z[(size_t)b * (2 * H1D) + f]       += mx;
  z[(size_t)b * (2 * H1D) + H1D + f] += sm / (float)n;
}

// out[row,:] = log_softmax(z2[row,:] @ W3 + b3)   -- one wave32 per row
__global__ void final_logsoftmax_kernel(const float* __restrict__ z2, const float* __restrict__ W3,
                                        const float* __restrict__ b3, float* __restrict__ out) {
  int row = blockIdx.x, lane = threadIdx.x;
  float logit = 0.f, v = -3.402823466e38f;
  if (lane < NCLS) {
    float s = b3[lane];
    const float* r = z2 + (size_t)row * H2D;
    for (int kk = 0; kk < H2D; ++kk) s += r[kk] * W3[kk * NCLS + lane];
    logit = s;
    v = s;
  }
  float mx = v;
  for (int d = 16; d >= 1; d >>= 1) mx = fmaxf(mx, __shfl_xor(mx, d, 32));
  float e = (lane < NCLS) ? expf(logit - mx) : 0.f;
  for (int d = 16; d >= 1; d >>= 1) e += __shfl_xor(e, d, 32);
  if (lane < NCLS) out[(size_t)row * NCLS + lane] = (logit - mx) - logf(e);
}

// ---------------------------------------------------------------------------
// Host orchestration
// ---------------------------------------------------------------------------
static inline int cdiv256(long long n) { return (int)((n + 255) / 256); }

extern "C" void kernel_launch(void* const* d_in, const int* in_sizes, int n_in,
                              void* d_out, int out_size, void* d_ws, size_t ws_size,
                              hipStream_t stream) {
  (void)in_sizes; (void)n_in; (void)out_size; (void)ws_size;
  const float* x     = (const float*)d_in[0];
  const int*   edges = (const int*)  d_in[1];
  const float* W12 = (const float*)d_in[2];  const float* b12 = (const float*)d_in[3];
  const float* W22 = (const float*)d_in[4];  const float* b22 = (const float*)d_in[5];
  const float* W32 = (const float*)d_in[6];  const float* b32 = (const float*)d_in[7];
  const float* pw1 = (const float*)d_in[8];
  const float* pw2 = (const float*)d_in[9];
  const float* pw3 = (const float*)d_in[10];
  const float* W1  = (const float*)d_in[11]; const float* b1 = (const float*)d_in[12];
  const float* W2  = (const float*)d_in[13]; const float* b2 = (const float*)d_in[14];
  const float* W3  = (const float*)d_in[15]; const float* b3 = (const float*)d_in[16];
  float* outp = (float*)d_out;

  // --- workspace carve-out ---
  char* w = (char*)d_ws;
  size_t off = 0;
  auto take = [&](size_t bytes) -> void* {
    void* p = w + off;
    off = (off + bytes + 255) & ~(size_t)255;
    return p;
  };
  const size_t HMAX = (size_t)BATCH * N0 * H1D;       // 8,388,608 floats
  float* hA    = (float*)take(HMAX * 4);
  float* hB    = (float*)take(HMAX * 4);
  float* deg   = (float*)take((size_t)BATCH * N0 * 4);    // reused as dinv in place
  float* enorm = (float*)take((size_t)BE * 4);
  float* score = (float*)take((size_t)BATCH * N0 * 4);
  float* pwn   = (float*)take(256);
  float* vals  = (float*)take((size_t)BATCH * N0 * 4);
  int*   perm  = (int*)  take((size_t)BATCH * N0 * 4);
  int*   inv   = (int*)  take((size_t)BATCH * N0 * 4);
  int*   srcA  = (int*)  take((size_t)BE * 4);
  int*   dstA  = (int*)  take((size_t)BE * 4);
  float* mA    = (float*)take((size_t)BE * 4);
  int*   srcB  = (int*)  take((size_t)BE * 4);
  int*   dstB  = (int*)  take((size_t)BE * 4);
  float* mB    = (float*)take((size_t)BE * 4);
  float* z     = (float*)take((size_t)BATCH * 512 * 4);
  float* z1    = (float*)take((size_t)BATCH * H1D * 4);
  float* z2    = (float*)take((size_t)BATCH * H2D * 4);

  auto gemm = [&](const float* A, const float* Wm, const float* bias, float* Cm,
                  int M, int N, int K, int relu) {
    int blocks = (M >> 7) * (N >> 4);                 // exact: M%128==0, N%16==0
    gemm_bias_act_wmma<<<blocks, 256, 0, stream>>>(A, Wm, bias, Cm, M, N, K, relu);
  };

  // K_PROP = 2 hops: h -> tmp -> h (result ends back in h)
  auto prop = [&](float* h, float* tmp, int n, const int* src, const int* dst, const float* m) {
    int Bn = BATCH * n;
    fill_f32_kernel<<<cdiv256(Bn), 256, 0, stream>>>(deg, 1.0f, Bn);
    deg_accum_kernel<<<cdiv256(BE), 256, 0, stream>>>(dst, m, deg, n, BE);
    dinv_kernel<<<cdiv256(Bn), 256, 0, stream>>>(deg, Bn);
    edge_norm_kernel<<<cdiv256(BE), 256, 0, stream>>>(src, dst, m, deg, enorm, n, BE);
    // hop 1: tmp = h*selfw + scatter(h)
    scale_self_kernel<<<cdiv256((long long)Bn * H1D), 256, 0, stream>>>(h, deg, tmp, Bn);
    edge_scatter_kernel<<<cdiv256((long long)BE * H1D), 256, 0, stream>>>(h, src, dst, enorm, tmp, n);
    // hop 2: h = tmp*selfw + scatter(tmp)
    scale_self_kernel<<<cdiv256((long long)Bn * H1D), 256, 0, stream>>>(tmp, deg, h, Bn);
    edge_scatter_kernel<<<cdiv256((long long)BE * H1D), 256, 0, stream>>>(tmp, src, dst, enorm, h, n);
  };

  auto pool = [&](const float* hin, float* hout, const float* pw, int n, int k,
                  const int* src, const int* dst, const float* m,
                  int* nsrc, int* ndst, float* nm) {
    int Bn = BATCH * n;
    pw_invnorm_kernel<<<1, 256, 0, stream>>>(pw, pwn);
    row_dot_kernel<<<(Bn + 7) / 8, 256, 0, stream>>>(hin, pw, pwn, score, Bn);
    topk_kernel<<<BATCH, 256, 0, stream>>>(score, vals, perm, n, k);
    (void)hipMemsetAsync(inv, 0xFF, (size_t)Bn * sizeof(int), stream);   // all -1
    pool_gather_kernel<<<BATCH * k, 256, 0, stream>>>(hin, vals, perm, inv, hout, n, k);
    edge_remap_kernel<<<cdiv256(BE), 256, 0, stream>>>(src, dst, m, inv, n, nsrc, ndst, nm, BE);
  };

  // ---- pipeline ----
  (void)hipMemsetAsync(z, 0, (size_t)BATCH * 512 * sizeof(float), stream);
  split_edges_kernel<<<cdiv256(BE), 256, 0, stream>>>(edges, srcA, dstA, mA, BE);

  // layer 1
  gemm(x, W12, b12, hA, BATCH * N0, H1D, F_IN, 1);            // 32768 x 128 x 256
  prop(hA, hB, N0, srcA, dstA, mA);
  pool(hA, hB, pw1, N0, KP1, srcA, dstA, mA, srcB, dstB, mB);
  readout_add_kernel<<<BATCH, H1D, 0, stream>>>(hB, z, KP1);

  // layer 2
  gemm(hB, W22, b22, hA, BATCH * KP1, H1D, H1D, 1);           // 26240 x 256 x 256
  prop(hA, hB, KP1, srcB, dstB, mB);
  pool(hA, hB, pw2, KP1, KP2, srcB, dstB, mB, srcA, dstA, mA);
  readout_add_kernel<<<BATCH, H1D, 0, stream>>>(hB, z, KP2);

  // layer 3
  gemm(hB, W32, b32, hA, BATCH * KP2, H1D, H1D, 1);           // 20992 x 256 x 256
  prop(hA, hB, KP2, srcA, dstA, mA);
  pool(hA, hB, pw3, KP2, KP3, srcA, dstA, mA, srcB, dstB, mB);
  readout_add_kernel<<<BATCH, H1D, 0, stream>>>(hB, z, KP3);

  // MLP head
  gemm(z,  W1, b1, z1, BATCH, H1D, 2 * H1D, 1);               // 128 x 512 x 256
  gemm(z1, W2, b2, z2, BATCH, H2D, H1D, 1);                   // 128 x 256 x 128
  final_logsoftmax_kernel<<<BATCH, 32, 0, stream>>>(z2, W3, b3, outp);
}